// SoftNMS3d_72610717106662
// MI455X (gfx1250) — compile-verified
//
#include <hip/hip_runtime.h>
#include <cstdint>
#include <cstddef>

#define DEV __device__ __forceinline__

// ---------------- problem constants (match reference) ----------------
constexpr int   Bn   = 16;
constexpr int   Cn   = 3;
constexpr int   Hd   = 1024;
constexpr int   Wd   = 1024;
constexpr int   HO   = 1022;   // valid conv, 3x3, pad 0, stride 1
constexpr int   WO   = 1022;
constexpr int   TILE = 32;     // output tile
constexpr int   TIN  = 34;     // input tile (halo)
constexpr int   LSTR = 36;     // LDS row stride (padding)
constexpr float K5L2E = 7.2134752044448169f; // 5 * log2(e)  -> exp(5x)=exp2(x*K)

// ---------------- CDNA5 async global->LDS staging ----------------
DEV void async_copy_b32(const float* gsrc, float* ldst) {
#if defined(__gfx1250__) && __has_builtin(__builtin_amdgcn_global_load_async_to_lds_b32)
    auto g = (__attribute__((address_space(1))) int*)(uintptr_t)gsrc;
    auto l = (__attribute__((address_space(3))) int*)(uint32_t)(uintptr_t)ldst;
    __builtin_amdgcn_global_load_async_to_lds_b32(g, l, 0, 0);
#else
    // fallback: per-lane async copy via raw ISA (VDST = LDS byte offset, VADDR = 64-bit global addr)
    uint32_t loff = (uint32_t)(uintptr_t)ldst;
    asm volatile("global_load_async_to_lds_b32 %0, %1, off"
                 :: "v"(loff), "v"(gsrc) : "memory");
#endif
}

DEV void wait_async0() {
#if defined(__gfx1250__) && __has_builtin(__builtin_amdgcn_s_wait_asynccnt)
    __builtin_amdgcn_s_wait_asynccnt(0);
#else
    asm volatile("s_wait_asynccnt 0" ::: "memory");
#endif
}

DEV float fast_exp2(float v) {
#if __has_builtin(__builtin_amdgcn_exp2f)
    return __builtin_amdgcn_exp2f(v);   // v_exp_f32
#else
    return exp2f(v);
#endif
}

// ---------------- fused SoftNMS3d kernel ----------------
__global__ __launch_bounds__(256)
void softnms3d_kernel(const float* __restrict__ x,
                      float* __restrict__ outResp,
                      float* __restrict__ outCoord,
                      float* __restrict__ outMask)
{
    __shared__ float raw[Cn][TIN][LSTR];  // staged inputs (async copy target)
    __shared__ float sE [TIN][LSTR];      // sum_c exp(5x)
    __shared__ float sEz[TIN][LSTR];      // e2 - e0   (z-offset weighted channel sum)
    __shared__ float sR [TIN][LSTR];      // sum_c exp(5x)*x

    const int tx0 = blockIdx.x * TILE;
    const int ty0 = blockIdx.y * TILE;
    const int b   = blockIdx.z;
    const int tid = threadIdx.x;

    // hint: pull the tile's first cachelines toward the WGP while we set up
    __builtin_prefetch(x + (((size_t)b * Cn) * Hd + ty0) * Wd + tx0, 0, 0);

    // ---- Stage A: async global -> LDS staging of 3 x 34 x 34 raw tile ----
    constexpr int nElem = Cn * TIN * TIN;  // 3468
    for (int i = tid; i < nElem; i += 256) {
        const int c  = i / (TIN * TIN);
        const int r  = (i / TIN) % TIN;
        const int cc = i % TIN;
        const int gy = ty0 + r;
        const int gx = tx0 + cc;
        if (gy < Hd && gx < Wd) {
            const float* gp = x + ((((size_t)b * Cn + c) * Hd + gy) * Wd + gx);
            async_copy_b32(gp, &raw[c][r][cc]);
        }
    }
    wait_async0();
    __syncthreads();

    // ---- Stage B: per-pixel channel reduction -> E, Ez, R planes ----
    for (int i = tid; i < TIN * TIN; i += 256) {
        const int r  = i / TIN;
        const int cc = i % TIN;
        const float x0 = raw[0][r][cc];
        const float x1 = raw[1][r][cc];
        const float x2 = raw[2][r][cc];
        const float e0 = fast_exp2(x0 * K5L2E);
        const float e1 = fast_exp2(x1 * K5L2E);
        const float e2 = fast_exp2(x2 * K5L2E);
        sE [r][cc] = e0 + e1 + e2;
        sEz[r][cc] = e2 - e0;                       // weights (c-1): -1,0,+1
        sR [r][cc] = e0 * x0 + e1 * x1 + e2 * x2;
    }
    __syncthreads();

    // ---- Stage C: 3x3 stencils -> 5 outputs per pixel ----
    const size_t planeO = (size_t)HO * WO;
#pragma unroll
    for (int k = 0; k < 4; ++k) {
        const int idx = tid + k * 256;  // 0..1023 over 32x32 tile
        const int ly  = idx >> 5;
        const int lx  = idx & 31;
        const int oy  = ty0 + ly;
        const int ox  = tx0 + lx;
        if (oy >= HO || ox >= WO) continue;

        float den = 0.f, nx = 0.f, ny = 0.f, nz = 0.f, nr = 0.f;
#pragma unroll
        for (int d = 0; d < 3; ++d) {
            const float a  = sE[ly + d][lx    ];
            const float m  = sE[ly + d][lx + 1];
            const float c2 = sE[ly + d][lx + 2];
            const float rowsum = a + m + c2;
            den += rowsum;
            nx  += (c2 - a);                         // dx weights -1,0,+1
            ny  += (float)(d - 1) * rowsum;          // dy weights -1,0,+1
            nz  += sEz[ly + d][lx] + sEz[ly + d][lx + 1] + sEz[ly + d][lx + 2];
            nr  += sR [ly + d][lx] + sR [ly + d][lx + 1] + sR [ly + d][lx + 2];
        }
        den += 1e-12f;
        const float inv  = 1.0f / den;
        const float resp = nr * inv;
        const float cz   = nz * inv;
        // NOTE: reference flips the global grid before the center-pick conv,
        // so the x-offset channel receives the *y* global coord and vice versa.
        const float c1   = nx * inv + (float)(oy + 1);
        const float c2v  = ny * inv + (float)(ox + 1);

        const size_t po = ((size_t)b * HO + oy) * WO + ox;
        const size_t pc = (((size_t)b * 3) * HO + oy) * (size_t)WO + ox;
        outResp[po]                = resp;
        outCoord[pc]               = cz;
        outCoord[pc + planeO]      = c1;
        outCoord[pc + 2 * planeO]  = c2v;
        outMask[po]                = (resp > 0.f) ? 1.0f : 0.0f;
    }
}

// ---------------- harness entry ----------------
extern "C" void kernel_launch(void* const* d_in, const int* in_sizes, int n_in,
                              void* d_out, int out_size, void* d_ws, size_t ws_size,
                              hipStream_t stream) {
    (void)in_sizes; (void)n_in; (void)d_ws; (void)ws_size; (void)out_size;
    const float* x = (const float*)d_in[0];
    float* out      = (float*)d_out;
    float* outResp  = out;                                  // (16,1,1022,1022)
    float* outCoord = out + (size_t)Bn * HO * WO;           // (16,3,1022,1022)
    float* outMask  = out + (size_t)Bn * HO * WO * 4;       // (16,1,1022,1022) as 0/1 float

    dim3 grid((WO + TILE - 1) / TILE, (HO + TILE - 1) / TILE, Bn);
    softnms3d_kernel<<<grid, 256, 0, stream>>>(x, outResp, outCoord, outMask);
}